// NonlinearFilterSmallL_68693706932921
// MI455X (gfx1250) — compile-verified
//
#include <hip/hip_runtime.h>
#include <hip/hip_bf16.h>
#include <math.h>

// Problem constants (from reference)
#define N_TRIALS 256
#define N_TIME   100
#define N_LAT    64
#define RANK     16
#define N_SAMP   32
#define HID      128

typedef float v2f __attribute__((ext_vector_type(2)));
typedef float v8f __attribute__((ext_vector_type(8)));

// ---------------------------------------------------------------------------
// One 16x16 fp32 WMMA tile: c += A(16xK) * B(Kx16), K stepped by 4 using
// V_WMMA_F32_16X16X4_F32.  A element (m,k) at pA[m*sAm + k*sAk];
// B element (k,n) at pB[k*sBk + n*sBn].  Lane layout per CDNA5 ISA 7.12.2:
//   A 16x4 : lane l -> m = l&15, k = k0 + 2*(l>>4) (+1 for .y)
//   B 4x16 : lane l -> n = l&15, k = k0 + 2*(l>>4) (+1 for .y)
//   C 16x16: VGPR i -> m = i + 8*(l>=16), n = l&15
// ---------------------------------------------------------------------------
__device__ __forceinline__ v8f wmma16x16_f32(const float* pA, int sAm, int sAk,
                                             const float* pB, int sBk, int sBn,
                                             int K, v8f c, int lane) {
  const int r = lane & 15;
  const int h = (lane >> 4) << 1;
  for (int k0 = 0; k0 < K; k0 += 4) {
    v2f a, b;
    a.x = pA[r * sAm + (k0 + h) * sAk];
    a.y = pA[r * sAm + (k0 + h + 1) * sAk];
    b.x = pB[(k0 + h) * sBk + r * sBn];
    b.y = pB[(k0 + h + 1) * sBk + r * sBn];
    c = __builtin_amdgcn_wmma_f32_16x16x4_f32(false, a, false, b, (short)0, c,
                                              false, false);
  }
  return c;
}

// C(MxN) = [C +] A(MxK)*B(KxN); C row-major (ldc) in LDS. 8 waves split tiles.
__device__ __forceinline__ void block_mm(const float* A, int sAm, int sAk,
                                         const float* B, int sBk, int sBn,
                                         float* C, int ldc,
                                         int M, int N, int K, bool acc, int tid) {
  const int lane = tid & 31, wave = tid >> 5;
  const int Tn = N >> 4;
  const int T = (M >> 4) * Tn;
  const int row = (lane >> 4) << 3;
  const int col = lane & 15;
  for (int tile = wave; tile < T; tile += 8) {
    const int tm = (tile / Tn) << 4;
    const int tn = (tile % Tn) << 4;
    v8f c = {};
    if (acc) {
#pragma unroll
      for (int i = 0; i < 8; ++i)
        c[i] = C[(tm + row + i) * ldc + tn + col];
    }
    c = wmma16x16_f32(A + (long)tm * sAm, sAm, sAk,
                      B + (long)tn * sBn, sBk, sBn, K, c, lane);
#pragma unroll
    for (int i = 0; i < 8; ++i)
      C[(tm + row + i) * ldc + tn + col] = c[i];
  }
}

// In-place Cholesky of 64x64 SPD matrix in LDS; lower factor kept, upper zeroed.
__device__ __forceinline__ void cholesky64(float* A, int tid) {
  __syncthreads();
  for (int k = 0; k < 64; ++k) {
    if (tid == 0) A[k * 64 + k] = sqrtf(fmaxf(A[k * 64 + k], 1e-20f));
    __syncthreads();
    float dk = A[k * 64 + k];
    if (tid > k && tid < 64) A[tid * 64 + k] /= dk;
    __syncthreads();
    for (int i = k + 1 + (tid >> 5); i < 64; i += 8) {
      float lik = A[i * 64 + k];
      for (int j = k + 1 + (tid & 31); j <= i; j += 32)
        A[i * 64 + j] -= lik * A[j * 64 + k];
    }
    __syncthreads();
  }
  for (int idx = tid; idx < 4096; idx += 256) {
    int i = idx >> 6, j = idx & 63;
    if (j > i) A[idx] = 0.0f;
  }
  __syncthreads();
}

// Li = L^{-1} for lower-triangular 64x64 L; one thread per column.
__device__ __forceinline__ void triinv64(const float* L, float* Li, int tid) {
  for (int idx = tid; idx < 4096; idx += 256) Li[idx] = 0.0f;
  __syncthreads();
  if (tid < 64) {
    const int j = tid;
    for (int i = j; i < 64; ++i) {
      float s = (i == j) ? 1.0f : 0.0f;
      for (int m = j; m < i; ++m) s -= L[i * 64 + m] * Li[m * 64 + j];
      Li[i * 64 + j] = s / L[i * 64 + i];
    }
  }
  __syncthreads();
}

// out = M v (or M^T v) for a 64x64 LDS matrix; threads 0..63 each one row/col.
__device__ __forceinline__ void mv64(const float* M, bool transp,
                                     const float* v, float* out, int tid) {
  if (tid < 64) {
    float acc = 0.0f;
    if (!transp) {
      for (int k = 0; k < 64; ++k) acc += M[tid * 64 + k] * v[k];
    } else {
      for (int k = 0; k < 64; ++k) acc += M[k * 64 + tid] * v[k];
    }
    out[tid] = acc;
  }
  __syncthreads();
}

__device__ __forceinline__ void write_outputs(const float* sMf, const float* sMp,
                                              const float* sPf, const float* sPp,
                                              float* o_mf, float* o_mp,
                                              float* o_Pf, float* o_Pp,
                                              int n, int t, int tid) {
  const long base = ((long)n * N_TIME + t) * N_LAT;
  if (tid < 64) {
    o_mf[base + tid] = sMf[tid];
    o_mp[base + tid] = sMp[tid];
  }
  const long baseM = base * N_LAT;
  for (int idx = tid; idx < 4096; idx += 256) {
    o_Pf[baseM + idx] = sPf[idx];
    o_Pp[baseM + idx] = sPp[idx];
  }
}

// KL: tr = ||Lp^-1 Pf||_F^2 (WMMA), qp = ||Lp^-1 (mf-mp)||^2, logdets from diags.
__device__ __forceinline__ void compute_kl(const float* sLpInv, const float* sPf,
                                           const float* sPp, const float* sMf,
                                           const float* sMp, float* sScr,
                                           float* sV1, float* sRed,
                                           float* o_kl, int n, int t, int tid) {
  block_mm(sLpInv, 64, 1, sPf, 64, 1, sScr, 64, 64, 64, 64, false, tid);
  if (tid < 64) {
    float acc = 0.0f;
    for (int k = 0; k < 64; ++k) acc += sLpInv[tid * 64 + k] * (sMf[k] - sMp[k]);
    sV1[tid] = acc;
  }
  __syncthreads();
  float part = 0.0f;
  for (int idx = tid; idx < 4096; idx += 256) {
    float v = sScr[idx];
    part += v * v;
  }
  sRed[tid] = part;
  __syncthreads();
  if (tid == 0) {
    float tr = 0.0f;
    for (int i = 0; i < 256; ++i) tr += sRed[i];
    float qp = 0.0f, ldp = 0.0f, ldf = 0.0f;
    for (int i = 0; i < 64; ++i) {
      qp  += sV1[i] * sV1[i];
      ldp += 2.0f * logf(fmaxf(sPp[i * 64 + i], 1e-30f));
      ldf += 2.0f * logf(fmaxf(sPf[i * 64 + i], 1e-30f));
    }
    o_kl[(long)n * N_TIME + t] = 0.5f * (tr + qp + ldp - ldf - (float)N_LAT);
  }
  __syncthreads();
}

// ---------------------------------------------------------------------------
// One workgroup per trial; 8 wave32s cooperate over the sequential scan.
// ---------------------------------------------------------------------------
__global__ __launch_bounds__(256)
void nlfilter_kernel(const float* __restrict__ kin, const float* __restrict__ Kin,
                     const float* __restrict__ w0in, const float* __restrict__ win,
                     const float* __restrict__ m0, const float* __restrict__ lq0,
                     const float* __restrict__ lq, const float* __restrict__ W1,
                     const float* __restrict__ b1, const float* __restrict__ W2,
                     const unsigned char* __restrict__ tmask,
                     float* __restrict__ out) {
  const int n = blockIdx.x;
  const int tid = threadIdx.x;

  float* o_z  = out;                                              // (S,N,T,L)
  float* o_mf = out + (long)N_SAMP * N_TRIALS * N_TIME * N_LAT;   // (N,T,L)
  float* o_mp = o_mf + (long)N_TRIALS * N_TIME * N_LAT;           // (N,T,L)
  float* o_Pf = o_mp + (long)N_TRIALS * N_TIME * N_LAT;           // (N,T,L,L)
  float* o_Pp = o_Pf + (long)N_TRIALS * N_TIME * N_LAT * N_LAT;   // (N,T,L,L)
  float* o_kl = o_Pp + (long)N_TRIALS * N_TIME * N_LAT * N_LAT;   // (N,T)

  __shared__ float sZ[N_SAMP * N_LAT];     // z samples [s][l]
  __shared__ float sH[N_SAMP * HID];       // MLP hidden; aliased as LJ^-1 later
  __shared__ float sMth[N_SAMP * N_LAT];   // mean_fn(z); also 64x32 scratch
  __shared__ float sPp[N_LAT * N_LAT];     // P_p -> Lp
  __shared__ float sLpInv[N_LAT * N_LAT];  // Lp^-1
  __shared__ float sJ[N_LAT * N_LAT];      // J_f -> LJ; KL scratch
  __shared__ float sPf[N_LAT * N_LAT];     // Pf = LJ^-T
  __shared__ float sWt[N_LAT * N_SAMP];    // w_t transposed [l][s]
  __shared__ float sKt[N_LAT * RANK];      // K_t [l][r]
  __shared__ float sMp[64], sMf[64], sV1[64], sV2[64], sP0[64], sQv[64];
  __shared__ float sRed[256];
  float* sLi = sH;  // LJ^-1 shares storage with MLP hidden activations

  if (tid < 64) {
    sP0[tid] = log1pf(expf(lq0[tid]));   // softplus
    sQv[tid] = log1pf(expf(lq[tid]));
  }
  __syncthreads();

  // ------------------------------- t = 0 ---------------------------------
  for (int idx = tid; idx < 4096; idx += 256) {
    int i = idx >> 6, j = idx & 63;
    sJ[idx] = (i == j) ? 1.0f / sP0[i] : 0.0f;          // diag(J0)
  }
  for (int idx = tid; idx < N_LAT * RANK; idx += 256)
    sKt[idx] = Kin[(long)n * N_TIME * (N_LAT * RANK) + idx];
  __syncthreads();
  block_mm(sKt, RANK, 1, sKt, 1, RANK, sJ, 64, 64, 64, RANK, true, tid);  // +K0 K0^T
  __syncthreads();
  cholesky64(sJ, tid);                  // LJ0
  triinv64(sJ, sLi, tid);               // LJ0^-1
  for (int idx = tid; idx < 4096; idx += 256) {
    int i = idx >> 6, j = idx & 63;
    sPf[idx] = sLi[j * 64 + i];         // Pf0 = LJ0^-T
  }
  for (int idx = tid; idx < 4096; idx += 256) {
    int i = idx >> 6, j = idx & 63;
    float d = sqrtf(sP0[i]);
    sPp[idx]    = (i == j) ? d : 0.0f;          // Pp0 = diag(sqrt(P0))
    sLpInv[idx] = (i == j) ? 1.0f / d : 0.0f;   // its inverse for KL
  }
  if (tid < 64) {
    sMp[tid] = m0[tid];
    sV2[tid] = m0[tid] / sP0[tid] + kin[(long)n * N_TIME * N_LAT + tid];
  }
  __syncthreads();
  mv64(sLi, false, sV2, sV1, tid);      // y  = LJ0^-1 rhs
  mv64(sLi, true,  sV1, sMf, tid);      // mf = LJ0^-T y
  for (int idx = tid; idx < N_LAT * N_SAMP; idx += 256) {
    int l = idx >> 5, s = idx & 31;
    sWt[idx] = w0in[((long)s * N_TRIALS + n) * N_LAT + l];
  }
  __syncthreads();
  block_mm(sPf, 64, 1, sWt, N_SAMP, 1, sMth, N_SAMP, 64, 32, 64, false, tid);
  __syncthreads();
  for (int idx = tid; idx < N_SAMP * N_LAT; idx += 256) {
    int s = idx >> 6, l = idx & 63;
    float zv = sMf[l] + sMth[l * N_SAMP + s];
    sZ[idx] = zv;
    o_z[((long)s * N_TRIALS + n) * N_TIME * N_LAT + l] = zv;   // t = 0
  }
  write_outputs(sMf, sMp, sPf, sPp, o_mf, o_mp, o_Pf, o_Pp, n, 0, tid);
  compute_kl(sLpInv, sPf, sPp, sMf, sMp, sJ, sV1, sRed, o_kl, n, 0, tid);
  __syncthreads();

  // ----------------------------- t = 1..T-1 -------------------------------
  for (int t = 1; t < N_TIME; ++t) {
    if (t + 1 < N_TIME && tid < 64) {   // prefetch next step (global_prefetch_b8)
      __builtin_prefetch(&Kin[(((long)n * N_TIME + t + 1) * N_LAT + tid) * RANK], 0, 1);
      __builtin_prefetch(&kin[((long)n * N_TIME + t + 1) * N_LAT + tid], 0, 1);
      __builtin_prefetch(&win[(((long)t * N_SAMP + (tid & 31)) * N_TRIALS + n) * N_LAT], 0, 1);
    }
    // --- mean_fn: m_th = z + tanh(z W1^T + b1) W2^T ---
    block_mm(sZ, 64, 1, W1, 1, 64, sH, HID, N_SAMP, HID, N_LAT, false, tid);
    __syncthreads();
    for (int idx = tid; idx < N_SAMP * HID; idx += 256)
      sH[idx] = tanhf(sH[idx] + b1[idx & 127]);
    __syncthreads();
    block_mm(sH, HID, 1, W2, 1, HID, sMth, 64, N_SAMP, N_LAT, HID, false, tid);
    __syncthreads();
    for (int idx = tid; idx < N_SAMP * N_LAT; idx += 256) sMth[idx] += sZ[idx];
    __syncthreads();
    // --- m_p and P_p = diag(Q) + Ezz - mp mp^T ---
    if (tid < 64) {
      float acc = 0.0f;
      for (int s = 0; s < N_SAMP; ++s) acc += sMth[s * 64 + tid];
      sMp[tid] = acc * (1.0f / N_SAMP);
    }
    __syncthreads();
    block_mm(sMth, 1, 64, sMth, 64, 1, sPp, 64, 64, 64, N_SAMP, false, tid);
    __syncthreads();
    for (int idx = tid; idx < 4096; idx += 256) {
      int i = idx >> 6, j = idx & 63;
      float v = sPp[idx] * (1.0f / N_SAMP) - sMp[i] * sMp[j];
      if (i == j) v += sQv[i];
      sPp[idx] = v;
    }
    cholesky64(sPp, tid);                 // Lp
    triinv64(sPp, sLpInv, tid);           // Lp^-1
    // --- J_f = Lp^-T Lp^-1 + K_t K_t^T ---
    block_mm(sLpInv, 1, 64, sLpInv, 64, 1, sJ, 64, 64, 64, 64, false, tid);
    for (int idx = tid; idx < N_LAT * RANK; idx += 256)
      sKt[idx] = Kin[((long)n * N_TIME + t) * (N_LAT * RANK) + idx];
    __syncthreads();
    block_mm(sKt, RANK, 1, sKt, 1, RANK, sJ, 64, 64, 64, RANK, true, tid);
    __syncthreads();
    cholesky64(sJ, tid);                  // LJ
    triinv64(sJ, sLi, tid);               // LJ^-1
    for (int idx = tid; idx < 4096; idx += 256) {
      int i = idx >> 6, j = idx & 63;
      sPf[idx] = sLi[j * 64 + i];         // Pf = LJ^-T
    }
    __syncthreads();
    // --- h_f = Lp^-T Lp^-1 m_p + k_t ;  m_f = LJ^-T LJ^-1 h_f ---
    mv64(sLpInv, false, sMp, sV1, tid);
    if (tid < 64) {
      float acc = 0.0f;
      for (int k = 0; k < 64; ++k) acc += sLpInv[k * 64 + tid] * sV1[k];
      sV2[tid] = acc + kin[((long)n * N_TIME + t) * N_LAT + tid];
    }
    __syncthreads();
    mv64(sLi, false, sV2, sV1, tid);
    mv64(sLi, true,  sV1, sMf, tid);
    // --- mask (block-uniform) ---
    if (tmask[t] != 0) {
      if (tid < 64) sMf[tid] = sMp[tid];
      for (int idx = tid; idx < 4096; idx += 256) sPf[idx] = sPp[idx];
    }
    __syncthreads();
    // --- z_f = m_f + Pf w_t ---
    for (int idx = tid; idx < N_LAT * N_SAMP; idx += 256) {
      int l = idx >> 5, s = idx & 31;
      sWt[idx] = win[(((long)(t - 1) * N_SAMP + s) * N_TRIALS + n) * N_LAT + l];
    }
    __syncthreads();
    block_mm(sPf, 64, 1, sWt, N_SAMP, 1, sMth, N_SAMP, 64, 32, 64, false, tid);
    __syncthreads();
    for (int idx = tid; idx < N_SAMP * N_LAT; idx += 256) {
      int s = idx >> 6, l = idx & 63;
      float zv = sMf[l] + sMth[l * N_SAMP + s];
      sZ[idx] = zv;
      o_z[(((long)s * N_TRIALS + n) * N_TIME + t) * N_LAT + l] = zv;
    }
    write_outputs(sMf, sMp, sPf, sPp, o_mf, o_mp, o_Pf, o_Pp, n, t, tid);
    compute_kl(sLpInv, sPf, sPp, sMf, sMp, sJ, sV1, sRed, o_kl, n, t, tid);
    __syncthreads();
  }
}

extern "C" void kernel_launch(void* const* d_in, const int* in_sizes, int n_in,
                              void* d_out, int out_size, void* d_ws, size_t ws_size,
                              hipStream_t stream) {
  const float* kin  = (const float*)d_in[0];
  const float* Kin  = (const float*)d_in[1];
  const float* w0in = (const float*)d_in[2];
  const float* win  = (const float*)d_in[3];
  const float* m0   = (const float*)d_in[4];
  const float* lq0  = (const float*)d_in[5];
  const float* lq   = (const float*)d_in[6];
  const float* W1   = (const float*)d_in[7];
  const float* b1   = (const float*)d_in[8];
  const float* W2   = (const float*)d_in[9];
  const unsigned char* tmask = (const unsigned char*)d_in[10];
  (void)in_sizes; (void)n_in; (void)d_ws; (void)ws_size; (void)out_size;
  nlfilter_kernel<<<N_TRIALS, 256, 0, stream>>>(kin, Kin, w0in, win, m0, lq0, lq,
                                                W1, b1, W2, tmask, (float*)d_out);
}